// BADDescriptor_82978768158930
// MI455X (gfx1250) — compile-verified
//
#include <hip/hip_runtime.h>

typedef __attribute__((ext_vector_type(2))) float v2f;
typedef __attribute__((ext_vector_type(4))) float v4f;
typedef __attribute__((ext_vector_type(8))) float v8f;

#define H_IMG 480
#define W_IMG 640
#define NPAIR 256

// ---------------------------------------------------------------------------
// Phase 1: 5x5 edge-clamped box average.
//   Horizontal 5-tap (VALU, clamped cols) for a 20x16 slab per tile -> LDS.
//   Vertical 5-tap as C(16x16) = sum_{j=0..4} A_j(16x4 selector) x B_j(4x16)
//   using V_WMMA_F32_16X16X4_F32 (wave32, exact: selector is 0/1, f32 acc).
//   One wave per 16x16 tile; 8 waves (8 tiles) per 256-thread block.
// ---------------------------------------------------------------------------
__global__ void __launch_bounds__(256)
box5_wmma_kernel(const float* __restrict__ x, float* __restrict__ box) {
  __shared__ float hext[8][20][16];   // per-wave 20-row x 16-col horizontal sums
  const int lane = threadIdx.x & 31;
  const int wv   = threadIdx.x >> 5;
  const int tile = blockIdx.x * 8 + wv;            // 1200 tiles total
  const int tyi  = tile / (W_IMG / 16);
  const int txi  = tile % (W_IMG / 16);
  const int r0 = tyi * 16, c0 = txi * 16;

  // Horizontal pass: rows r0-2..r0+17 (clamped), cols c0..c0+15, 5-tap clamped.
  for (int e = lane; e < 20 * 16; e += 32) {
    int kk = e >> 4;         // 0..19 slab row
    int n  = e & 15;         // 0..15 col within tile
    int row = r0 - 2 + kk;
    row = row < 0 ? 0 : (row > H_IMG - 1 ? H_IMG - 1 : row);
    const float* rp = x + row * W_IMG;
    float s = 0.0f;
#pragma unroll
    for (int d = -2; d <= 2; ++d) {
      int col = c0 + n + d;
      col = col < 0 ? 0 : (col > W_IMG - 1 ? W_IMG - 1 : col);
      s += rp[col];
    }
    hext[wv][kk][n] = s;
  }
  __syncthreads();

  const int hi = lane >> 4;   // lane half: selects K pair / M offset
  const int mn = lane & 15;   // M for A, N for B & C-store

  v8f c = {};
#if __has_builtin(__builtin_amdgcn_wmma_f32_16x16x4_f32)
#pragma unroll
  for (int j = 0; j < 5; ++j) {
    const int k0 = 4 * j + 2 * hi;   // A/B layout: vgpr0 -> K = 2*hi, vgpr1 -> K = 2*hi+1
    const int k1 = k0 + 1;
    v2f a, b;
    // Selector: out row m needs slab rows kk in [m, m+4]
    a[0] = (k0 >= mn && k0 <= mn + 4) ? 1.0f : 0.0f;
    a[1] = (k1 >= mn && k1 <= mn + 4) ? 1.0f : 0.0f;
    b[0] = hext[wv][k0][mn];
    b[1] = hext[wv][k1][mn];
    c = __builtin_amdgcn_wmma_f32_16x16x4_f32(false, a, false, b, (short)0, c,
                                              false, false);
  }
#else
  // Fallback: plain VALU vertical 5-tap (C layout emulated per lane)
#pragma unroll
  for (int r = 0; r < 8; ++r) {
    int m = r + 8 * hi;
    float s = 0.0f;
#pragma unroll
    for (int kk = 0; kk < 5; ++kk) s += hext[wv][m + kk][mn];
    c[r] = s;
  }
#endif

  // Store C (vgpr r: M = r + 8*hi, N = mn), scaled by 1/25.
#pragma unroll
  for (int r = 0; r < 8; ++r) {
    int row = r0 + r + 8 * hi;
    box[row * W_IMG + c0 + mn] = c[r] * 0.04f;
  }
}

// ---------------------------------------------------------------------------
// Phase 2: per-pair translational bilinear sampling + difference.
//   Store-bandwidth bound: 314.6 MB streamed out -> float4 non-temporal
//   stores; 8 reads/element hit the L2-resident 1.2 MB box buffer and are
//   coalesced (constant per-pair shift). y-terms hoisted per 4-pixel group.
// ---------------------------------------------------------------------------
__global__ void __launch_bounds__(256)
bad_sample_kernel(const float* __restrict__ box, const float* __restrict__ offs,
                  float* __restrict__ out) {
  const int p    = blockIdx.y;
  const int g    = blockIdx.x * blockDim.x + threadIdx.x;
  const int base = g * 4;                 // 4 consecutive pixels in one row
  const int y    = base / W_IMG;
  const int x0   = base - y * W_IMG;

  // Mirror reference math: eff = off * (2/(dim-1+1e-8)) * 0.5 * (dim-1)
  const float sy = 0.5f * (float)(H_IMG - 1) * (2.0f / ((float)(H_IMG - 1) + 1e-8f));
  const float sx = 0.5f * (float)(W_IMG - 1) * (2.0f / ((float)(W_IMG - 1) + 1e-8f));

  const v4f o = *(const v4f*)(offs + p * 4);  // (dy1, dx1, dy2, dx2)
  const float ey1 = o[0] * sy, ex1 = o[1] * sx;
  const float ey2 = o[2] * sy, ex2 = o[3] * sx;

  // y-dependent terms (shared by the 4 pixels of this lane)
  float py1 = fminf(fmaxf((float)y + ey1, 0.0f), (float)(H_IMG - 1));
  float py2 = fminf(fmaxf((float)y + ey2, 0.0f), (float)(H_IMG - 1));
  float fy1 = floorf(py1), fy2 = floorf(py2);
  float wy1 = py1 - fy1,   wy2 = py2 - fy2;
  int iy1a = (int)fy1; int iy1b = min(iy1a + 1, H_IMG - 1);
  int iy2a = (int)fy2; int iy2b = min(iy2a + 1, H_IMG - 1);
  const float* r1a = box + iy1a * W_IMG;
  const float* r1b = box + iy1b * W_IMG;
  const float* r2a = box + iy2a * W_IMG;
  const float* r2b = box + iy2b * W_IMG;

  v4f res;
#pragma unroll
  for (int i = 0; i < 4; ++i) {
    float xf = (float)(x0 + i);
    float px1 = fminf(fmaxf(xf + ex1, 0.0f), (float)(W_IMG - 1));
    float px2 = fminf(fmaxf(xf + ex2, 0.0f), (float)(W_IMG - 1));
    float fx1 = floorf(px1), fx2 = floorf(px2);
    float wx1 = px1 - fx1,   wx2 = px2 - fx2;
    int ix1a = (int)fx1; int ix1b = min(ix1a + 1, W_IMG - 1);
    int ix2a = (int)fx2; int ix2b = min(ix2a + 1, W_IMG - 1);
    float v1 = (1.0f - wy1) * ((1.0f - wx1) * r1a[ix1a] + wx1 * r1a[ix1b]) +
                       wy1  * ((1.0f - wx1) * r1b[ix1a] + wx1 * r1b[ix1b]);
    float v2 = (1.0f - wy2) * ((1.0f - wx2) * r2a[ix2a] + wx2 * r2a[ix2b]) +
                       wy2  * ((1.0f - wx2) * r2b[ix2a] + wx2 * r2b[ix2b]);
    res[i] = v1 - v2;
  }

  v4f* dst = (v4f*)(out + ((size_t)p * (H_IMG * W_IMG) + base));
  __builtin_nontemporal_store(res, dst);   // streaming 314.6 MB: keep L2 for box
}

// ---------------------------------------------------------------------------
extern "C" void kernel_launch(void* const* d_in, const int* in_sizes, int n_in,
                              void* d_out, int out_size, void* d_ws, size_t ws_size,
                              hipStream_t stream) {
  const float* x    = (const float*)d_in[0];   // [1,1,480,640]
  const float* offs = (const float*)d_in[1];   // [256,2,2] -> (dy1,dx1,dy2,dx2) per pair
  float* out = (float*)d_out;                  // [1,256,480,640]
  float* box = (float*)d_ws;                   // scratch: 480*640*4 = 1.2 MB

  // Phase 1: 1200 tiles of 16x16, 8 waves/block -> 150 blocks.
  box5_wmma_kernel<<<dim3((H_IMG / 16) * (W_IMG / 16) / 8), 256, 0, stream>>>(x, box);

  // Phase 2: 4 px/thread -> 300 blocks along pixels, 256 pairs on y.
  dim3 grid2((H_IMG * W_IMG) / (256 * 4), NPAIR);
  bad_sample_kernel<<<grid2, 256, 0, stream>>>(box, offs, out);
}